// ResD1LayerV2_6176162972422
// MI455X (gfx1250) — compile-verified
//
#include <hip/hip_runtime.h>
#include <math.h>
#include <stdint.h>

// Problem dims (fixed by reference)
#define NB 1024   // batch rows
#define ND 128    // input dim
#define NH 128    // hidden dim
#define LN_EPS 1e-5f

typedef float v2f __attribute__((ext_vector_type(2)));
typedef float v8f __attribute__((ext_vector_type(8)));

// ---------------------------------------------------------------------------
// CDNA5 async global->LDS DMA (tracked by ASYNCcnt). Per-lane 16B copy.
// LDS generic pointers carry the LDS byte offset in their low 32 bits
// (ISA 10.2 aperture rules), so the cast yields the LDS-address VGPR operand.
// ---------------------------------------------------------------------------
__device__ __forceinline__ void async_copy_b128(float* lds_dst, const float* gsrc) {
  unsigned laddr = (unsigned)(uintptr_t)lds_dst;
  asm volatile("global_load_async_to_lds_b128 %0, %1, off"
               :: "v"(laddr), "v"((const float4*)gsrc)
               : "memory");
}

__device__ __forceinline__ void async_wait0() {
  asm volatile("s_wait_asynccnt 0" ::: "memory");
}

// Stage n_f4 float4s cooperatively (256 threads) via the async DMA path.
__device__ __forceinline__ void stage_async(float* lds, const float* gsrc,
                                            int n_f4, int tid) {
  for (int idx = tid; idx < n_f4; idx += 256)
    async_copy_b128(lds + idx * 4, gsrc + idx * 4);
}

// ---------------------------------------------------------------------------
// Kernel 1: diffs_mean[i,d] = (1/B) * sum_j |x[i,d] - x[j,d]|
// Block = 256 thr = 8 waves. Wave w owns dim-group [16w,16w+16); lane owns row
// i = blockIdx.x*32 + lane. j streams through 2 x 32KB LDS tiles, loaded with
// async-to-LDS DMA and double-buffered against the compute on the prior tile.
// Inner LDS reads are lane-uniform (hardware broadcast, no bank conflicts).
// ---------------------------------------------------------------------------
__global__ void k_pairdiff(const float* __restrict__ x,
                           float* __restrict__ dm) {
  __shared__ __align__(16) float xj[2][64 * 128];  // 2 x 32 KB j-tiles
  const int tid  = threadIdx.x;
  const int lane = tid & 31;
  const int wv   = tid >> 5;           // 0..7 dim-group
  const int i    = blockIdx.x * 32 + lane;
  const int db   = wv * 16;

  float xi[16], acc[16];
#pragma unroll
  for (int k = 0; k < 16; ++k) {
    xi[k]  = x[i * ND + db + k];
    acc[k] = 0.0f;
  }

  const int NT = NB / 64;  // 16 tiles of 64 j-rows
  // prefetch tile 0
  stage_async(&xj[0][0], x, 64 * 32, tid);
  async_wait0();
  __syncthreads();

  for (int t = 0; t < NT; ++t) {
    if (t + 1 < NT)  // overlap: stream tile t+1 while computing on tile t
      stage_async(&xj[(t + 1) & 1][0], x + (t + 1) * 64 * ND, 64 * 32, tid);

    const float* buf = &xj[t & 1][0];
#pragma unroll 4
    for (int jj = 0; jj < 64; ++jj) {
      const float* row = &buf[jj * 128 + db];  // lane-uniform address
#pragma unroll
      for (int k = 0; k < 16; ++k) acc[k] += fabsf(xi[k] - row[k]);
    }

    async_wait0();     // my async loads for tile t+1 are done
    __syncthreads();   // everyone's are done; tile t fully consumed
  }

  const float s = 1.0f / (float)NB;
#pragma unroll
  for (int k = 0; k < 16; ++k) dm[i * ND + db + k] = acc[k] * s;
}

// ---------------------------------------------------------------------------
// Kernel 2: tau[i] = clamp(softplus(x[i]·Wt + bt), 0.01, inf) + 1
// One wave per row; 4 elems per lane; shuffle reduction.
// ---------------------------------------------------------------------------
__global__ void k_tau(const float* __restrict__ x,
                      const float* __restrict__ Wt,
                      const float* __restrict__ bt,
                      float* __restrict__ tau) {
  const int lane = threadIdx.x & 31;
  const int wv   = threadIdx.x >> 5;
  const int row  = blockIdx.x * 8 + wv;
  float p = 0.0f;
#pragma unroll
  for (int k = 0; k < 4; ++k) {
    const int d = lane + 32 * k;
    p += x[row * ND + d] * Wt[d];
  }
#pragma unroll
  for (int m = 16; m >= 1; m >>= 1) p += __shfl_xor(p, m, 32);
  if (lane == 0) {
    const float z  = p + bt[0];
    const float sp = (z > 20.0f) ? z : log1pf(expf(z));  // softplus
    tau[row] = fmaxf(sp, 0.01f) + 1.0f;
  }
}

// ---------------------------------------------------------------------------
// fp32 WMMA 16x16x4 tile GEMM: one wave computes C[16 x 128] = A[16 x 128] x W^T
// A read directly from global (L2-resident), W staged in LDS (row-major W[n][k]).
// A-layout (ISA 7.12.2): lane = (half, lm): a.v = A[M=lm][K = k0 + v + 2*half]
// B-layout mirrored:                        b.v = W[N=lm][K = k0 + v + 2*half]
// C-layout: acc[nt][v] = C[M = v + 8*half][N = nt*16 + lm]
// ---------------------------------------------------------------------------
__device__ __forceinline__ void wmma_tile_gemm(const float* __restrict__ Ag,
                                               const float* __restrict__ Wl,
                                               v8f acc[8]) {
  const int lane = threadIdx.x & 31;
  const int half = lane >> 4;
  const int lm   = lane & 15;
  for (int k0 = 0; k0 < 128; k0 += 4) {
    const int ka = k0 + 2 * half;
    v2f a;
    a.x = Ag[lm * 128 + ka];
    a.y = Ag[lm * 128 + ka + 1];
#pragma unroll
    for (int nt = 0; nt < 8; ++nt) {
      v2f b;
      b.x = Wl[(nt * 16 + lm) * 128 + ka];
      b.y = Wl[(nt * 16 + lm) * 128 + ka + 1];
      acc[nt] = __builtin_amdgcn_wmma_f32_16x16x4_f32(
          false, a, false, b, (short)0, acc[nt], false, false);
    }
  }
}

// ---------------------------------------------------------------------------
// Kernel 3: A2 = (diffs_mean @ Wd^T + bd) / tau     [1024 x 128]
// Block = 128 rows; wave w owns m-tile w. Weights staged via async DMA.
// ---------------------------------------------------------------------------
__global__ void k_gemm1(const float* __restrict__ dm,
                        const float* __restrict__ Wd,
                        const float* __restrict__ bd,
                        const float* __restrict__ tau,
                        float* __restrict__ A2) {
  __shared__ __align__(16) float Wl[128 * 128];  // 64 KB
  const int tid = threadIdx.x;
  stage_async(Wl, Wd, 128 * 32, tid);
  async_wait0();
  __syncthreads();

  const int wv      = tid >> 5;
  const int lane    = tid & 31;
  const int half    = lane >> 4;
  const int lm      = lane & 15;
  const int rowBase = blockIdx.x * 128 + wv * 16;

  v8f acc[8];
  v8f zero = {};
#pragma unroll
  for (int nt = 0; nt < 8; ++nt) acc[nt] = zero;

  wmma_tile_gemm(dm + rowBase * ND, Wl, acc);

  float rtau[8];
#pragma unroll
  for (int v = 0; v < 8; ++v) rtau[v] = 1.0f / tau[rowBase + v + 8 * half];

#pragma unroll
  for (int nt = 0; nt < 8; ++nt) {
    const int n  = nt * 16 + lm;
    const float bn = bd[n];
#pragma unroll
    for (int v = 0; v < 8; ++v) {
      const int m = rowBase + v + 8 * half;
      A2[m * NH + n] = (acc[nt][v] + bn) * rtau[v];
    }
  }
}

// ---------------------------------------------------------------------------
// Kernel 4: y = relu(A2 @ Wa^T + ba) + x @ Wr^T + br ; out = LayerNorm(y)
// Two WMMA passes time-multiplexing one 64KB LDS weight buffer (async staged).
// LayerNorm fully in-register: row m lives in the 16 lanes of one half, so
// shfl_xor masks 1/2/4/8 reduce exactly over one row.
// ---------------------------------------------------------------------------
__global__ void k_fused(const float* __restrict__ A2,
                        const float* __restrict__ Wa,
                        const float* __restrict__ ba,
                        const float* __restrict__ x,
                        const float* __restrict__ Wr,
                        const float* __restrict__ br,
                        const float* __restrict__ gamma,
                        const float* __restrict__ beta,
                        float* __restrict__ out) {
  __shared__ __align__(16) float Wl[128 * 128];  // 64 KB, reused Wa -> Wr
  const int tid     = threadIdx.x;
  const int wv      = tid >> 5;
  const int lane    = tid & 31;
  const int half    = lane >> 4;
  const int lm      = lane & 15;
  const int rowBase = blockIdx.x * 128 + wv * 16;

  v8f accH[8], accR[8];
  v8f zero = {};
#pragma unroll
  for (int nt = 0; nt < 8; ++nt) { accH[nt] = zero; accR[nt] = zero; }

  // Pass 1: h-pre = A2 @ Wa^T
  stage_async(Wl, Wa, 128 * 32, tid);
  async_wait0();
  __syncthreads();
  wmma_tile_gemm(A2 + rowBase * NH, Wl, accH);
  __syncthreads();

  // Pass 2: r = x @ Wr^T
  stage_async(Wl, Wr, 128 * 32, tid);
  async_wait0();
  __syncthreads();
  wmma_tile_gemm(x + rowBase * ND, Wl, accR);

  // y = relu(accH + ba) + accR + br ; accumulate row stats
  float rs[8], rq[8];
#pragma unroll
  for (int v = 0; v < 8; ++v) { rs[v] = 0.0f; rq[v] = 0.0f; }

#pragma unroll
  for (int nt = 0; nt < 8; ++nt) {
    const int n   = nt * 16 + lm;
    const float ban = ba[n];
    const float brn = br[n];
#pragma unroll
    for (int v = 0; v < 8; ++v) {
      const float yv = fmaxf(accH[nt][v] + ban, 0.0f) + accR[nt][v] + brn;
      accH[nt][v] = yv;  // stash y
      rs[v] += yv;
      rq[v] += yv * yv;
    }
  }

  // reduce over the 16 lanes of this half (one row per v)
#pragma unroll
  for (int m = 1; m < 16; m <<= 1) {
#pragma unroll
    for (int v = 0; v < 8; ++v) {
      rs[v] += __shfl_xor(rs[v], m, 32);
      rq[v] += __shfl_xor(rq[v], m, 32);
    }
  }

  float mu[8], rstd[8];
  const float invH = 1.0f / (float)NH;
#pragma unroll
  for (int v = 0; v < 8; ++v) {
    mu[v] = rs[v] * invH;
    const float var = rq[v] * invH - mu[v] * mu[v];
    rstd[v] = rsqrtf(var + LN_EPS);
  }

#pragma unroll
  for (int nt = 0; nt < 8; ++nt) {
    const int n   = nt * 16 + lm;
    const float g = gamma[n];
    const float bt = beta[n];
#pragma unroll
    for (int v = 0; v < 8; ++v) {
      const int m = rowBase + v + 8 * half;
      out[m * NH + n] = (accH[nt][v] - mu[v]) * rstd[v] * g + bt;
    }
  }
}

// ---------------------------------------------------------------------------
// Host launcher
// d_in order: x, Wd, bd, Wt, bt, Wa, ba, Wr, br, gamma, beta
// ---------------------------------------------------------------------------
extern "C" void kernel_launch(void* const* d_in, const int* in_sizes, int n_in,
                              void* d_out, int out_size, void* d_ws, size_t ws_size,
                              hipStream_t stream) {
  const float* x     = (const float*)d_in[0];
  const float* Wd    = (const float*)d_in[1];
  const float* bd    = (const float*)d_in[2];
  const float* Wt    = (const float*)d_in[3];
  const float* bt    = (const float*)d_in[4];
  const float* Wa    = (const float*)d_in[5];
  const float* ba    = (const float*)d_in[6];
  const float* Wr    = (const float*)d_in[7];
  const float* br    = (const float*)d_in[8];
  const float* gamma = (const float*)d_in[9];
  const float* beta  = (const float*)d_in[10];
  float* out = (float*)d_out;

  // workspace layout
  char* ws = (char*)d_ws;
  float* dm  = (float*)(ws);                            // 1024*128 f32 = 512 KB
  float* A2  = (float*)(ws + (size_t)NB * ND * 4);      // 512 KB
  float* tau = (float*)(ws + (size_t)2 * NB * ND * 4);  // 4 KB

  k_pairdiff<<<NB / 32, 256, 0, stream>>>(x, dm);
  k_tau<<<NB / 8, 256, 0, stream>>>(x, Wt, bt, tau);
  k_gemm1<<<NB / 128, 256, 0, stream>>>(dm, Wd, bd, tau, A2);
  k_fused<<<NB / 128, 256, 0, stream>>>(A2, Wa, ba, x, Wr, br, gamma, beta, out);
}